// AdditiveAttention_46634754900130
// MI455X (gfx1250) — compile-verified
//
#include <hip/hip_runtime.h>
#include <hip/hip_bf16.h>

// Problem constants (from reference)
#define B_ 4
#define Q_ 512
#define K_ 512
#define D_ 256   // KEY_SIZE == QUERY_SIZE
#define H_ 128   // HIDDEN
#define V_ 256   // VDIM

typedef float v2f __attribute__((ext_vector_type(2)));
typedef float v8f __attribute__((ext_vector_type(8)));

// ---------- fast transcendentals (gfx1250 TRANS pipe) ----------
__device__ __forceinline__ float fast_tanh(float x) {
#if __has_builtin(__builtin_amdgcn_tanhf)
  return __builtin_amdgcn_tanhf(x);
#elif __has_builtin(__builtin_amdgcn_tanh_f32)
  return __builtin_amdgcn_tanh_f32(x);
#else
  float r;
  // v_nop covers the TRANS-op result-use hazard (compiler can't see through asm)
  asm("v_tanh_f32 %0, %1\n\tv_nop" : "=v"(r) : "v"(x));
  return r;
#endif
}

__device__ __forceinline__ float fast_exp(float x) {
  const float L2E = 1.44269504088896340736f;
#if __has_builtin(__builtin_amdgcn_exp2f)
  return __builtin_amdgcn_exp2f(x * L2E);
#else
  float r, xs = x * L2E;
  asm("v_exp_f32 %0, %1\n\tv_nop" : "=v"(r) : "v"(xs));
  return r;
#endif
}

// ---------- Kernel 1: projections qp = query@Wq^T, kp = key@Wk^T ------------
// One wave per 16x16 output tile; K-loop of V_WMMA_F32_16X16X4_F32.
// 128 row tiles (2048 rows, B*Q == B*K flattened) x 8 column tiles x 2 projections = 2048 waves.
__global__ __launch_bounds__(128) void proj_kernel(
    const float* __restrict__ query, const float* __restrict__ key,
    const float* __restrict__ Wq, const float* __restrict__ Wk,
    float* __restrict__ qp, float* __restrict__ kp) {
  const int lane = threadIdx.x & 31;
  const int w = blockIdx.x * 4 + (threadIdx.x >> 5);  // 0..2047
  const int sel = w >> 10;                            // 0: q-proj, 1: k-proj
  const int t = w & 1023;
  const int row0 = (t >> 3) * 16;                     // flattened B*rows
  const int h0 = (t & 7) * 16;

  const float* X = sel ? key : query;
  const float* W = sel ? Wk : Wq;
  float* O = sel ? kp : qp;

  // ISA 7.12.2 layouts for 32-bit 16x4 A / 4x16 B
  const int i = lane & 15;           // A row M / B col N
  const int kb = (lane >> 4) << 1;   // K base: 0 (lanes 0-15) or 2 (16-31)

  const float* arow = X + (size_t)(row0 + i) * D_;   // A[i][k]
  const float* wrow = W + (size_t)(h0 + i) * D_;     // B[k][n] = W[h0+n][k]; n==i

  v8f acc = {};
#pragma unroll 4
  for (int k0 = 0; k0 < D_; k0 += 4) {
    v2f a, b;
    a.x = arow[k0 + kb];
    a.y = arow[k0 + kb + 1];
    b.x = wrow[k0 + kb];
    b.y = wrow[k0 + kb + 1];
    acc = __builtin_amdgcn_wmma_f32_16x16x4_f32(false, a, false, b, (short)0,
                                                acc, false, false);
  }
  const int mo = (lane >> 4) << 3;  // C/D: VGPR r -> M = r + 8*(lane/16)
#pragma unroll
  for (int r = 0; r < 8; ++r)
    O[(size_t)(row0 + mo + r) * H_ + h0 + i] = acc[r];
}

// ---------- Kernel 2: additive scores with hw tanh --------------------------
// Block = 256 threads handles a (16 q) x (64 k) tile of one batch.
// LDS rows padded to 129 words -> conflict-free across the 64 banks.
#define QS_STRIDE 129
#define KS_STRIDE 129
__global__ __launch_bounds__(256) void score_kernel(
    const float* __restrict__ qp, const float* __restrict__ kp,
    const float* __restrict__ Wv, const unsigned char* __restrict__ mask,
    float* __restrict__ scores) {
  __shared__ float qs[16 * QS_STRIDE];
  __shared__ float ks[64 * KS_STRIDE];
  __shared__ float wv[H_];

  const int b = blockIdx.z;
  const int q0 = blockIdx.y * 16;
  const int k0 = blockIdx.x * 64;
  const int t = threadIdx.x;

  const float* qsrc = qp + ((size_t)b * Q_ + q0) * H_;  // 16x128, contiguous
  const float* ksrc = kp + ((size_t)b * K_ + k0) * H_;  // 64x128, contiguous

  for (int idx = t; idx < 16 * H_; idx += 256)
    qs[(idx >> 7) * QS_STRIDE + (idx & 127)] = qsrc[idx];
  for (int idx = t; idx < 64 * H_; idx += 256)
    ks[(idx >> 7) * KS_STRIDE + (idx & 127)] = ksrc[idx];
  if (t < H_) wv[t] = Wv[t];
  __syncthreads();

  const int i = t & 15;          // q row within tile
  const int j0 = (t >> 4) * 4;   // first of 4 k columns
  const float* qrow = qs + i * QS_STRIDE;
  const float* kr0 = ks + (j0 + 0) * KS_STRIDE;
  const float* kr1 = ks + (j0 + 1) * KS_STRIDE;
  const float* kr2 = ks + (j0 + 2) * KS_STRIDE;
  const float* kr3 = ks + (j0 + 3) * KS_STRIDE;

  float a0 = 0.f, a1 = 0.f, a2 = 0.f, a3 = 0.f;
#pragma unroll 4
  for (int h = 0; h < H_; ++h) {
    const float qv = qrow[h];
    const float w = wv[h];
    a0 += w * fast_tanh(qv + kr0[h]);
    a1 += w * fast_tanh(qv + kr1[h]);
    a2 += w * fast_tanh(qv + kr2[h]);
    a3 += w * fast_tanh(qv + kr3[h]);
  }

  const size_t base = ((size_t)b * Q_ + q0 + i) * K_ + k0 + j0;
  const unsigned char* mrow = mask + base;
  float* srow = scores + base;
  const float ninf = -__builtin_inff();
  srow[0] = mrow[0] ? ninf : a0;
  srow[1] = mrow[1] ? ninf : a1;
  srow[2] = mrow[2] ? ninf : a2;
  srow[3] = mrow[3] ? ninf : a3;
}

// ---------- Kernel 3: softmax over K (in place) -----------------------------
__global__ __launch_bounds__(256) void softmax_kernel(float* __restrict__ scores) {
  __shared__ float red[256];
  float* s = scores + (size_t)blockIdx.x * K_;
  const int t = threadIdx.x;

  float x0 = s[t], x1 = s[t + 256];
  red[t] = fmaxf(x0, x1);
  __syncthreads();
  for (int off = 128; off > 0; off >>= 1) {
    if (t < off) red[t] = fmaxf(red[t], red[t + off]);
    __syncthreads();
  }
  const float m = red[0];
  __syncthreads();

  const float e0 = fast_exp(x0 - m);
  const float e1 = fast_exp(x1 - m);
  red[t] = e0 + e1;
  __syncthreads();
  for (int off = 128; off > 0; off >>= 1) {
    if (t < off) red[t] += red[t + off];
    __syncthreads();
  }
  const float inv = 1.0f / red[0];
  s[t] = e0 * inv;
  s[t + 256] = e1 * inv;
}

// ---------- Kernel 4: out = attn @ value via f32 WMMA -----------------------
// One wave per 16x16 output tile; 128 WMMAs along K=512.
__global__ __launch_bounds__(128) void av_kernel(
    const float* __restrict__ attn, const float* __restrict__ value,
    float* __restrict__ out) {
  const int lane = threadIdx.x & 31;
  const int w = blockIdx.x * 4 + (threadIdx.x >> 5);  // 0..2047
  const int b = w >> 9;                               // 512 tiles per batch
  const int tt = w & 511;
  const int m0 = (tt >> 4) * 16;                      // q-tile
  const int n0 = (tt & 15) * 16;                      // v-tile

  const int i = lane & 15;
  const int kb = (lane >> 4) << 1;

  const float* arow = attn + ((size_t)b * Q_ + m0 + i) * K_;
  const float* vcol = value + (size_t)b * K_ * V_ + n0 + i;  // B[k][n], n==i

  v8f acc = {};
#pragma unroll 4
  for (int kk = 0; kk < K_; kk += 4) {
    v2f a, bb;
    a.x = arow[kk + kb];
    a.y = arow[kk + kb + 1];
    bb.x = vcol[(size_t)(kk + kb) * V_];
    bb.y = vcol[(size_t)(kk + kb + 1) * V_];
    acc = __builtin_amdgcn_wmma_f32_16x16x4_f32(false, a, false, bb, (short)0,
                                                acc, false, false);
  }
  const int mo = (lane >> 4) << 3;
#pragma unroll
  for (int r = 0; r < 8; ++r)
    out[((size_t)b * Q_ + m0 + mo + r) * V_ + n0 + i] = acc[r];
}

// ---------- launch ----------------------------------------------------------
extern "C" void kernel_launch(void* const* d_in, const int* in_sizes, int n_in,
                              void* d_out, int out_size, void* d_ws, size_t ws_size,
                              hipStream_t stream) {
  const float* key = (const float*)d_in[0];           // [B,K,256]
  const float* query = (const float*)d_in[1];         // [B,Q,256]
  const float* value = (const float*)d_in[2];         // [B,K,256]
  const unsigned char* mask = (const unsigned char*)d_in[3];  // [B,Q,K] bool
  const float* Wk = (const float*)d_in[4];            // [128,256]
  const float* Wq = (const float*)d_in[5];            // [128,256]
  const float* Wv = (const float*)d_in[6];            // [128]
  float* out = (float*)d_out;                         // [B,Q,256]

  // workspace: qp (1MB) | kp (1MB) | scores (4MB) = 6MB total
  float* qp = (float*)d_ws;
  float* kp = qp + (size_t)B_ * Q_ * H_;
  float* sc = kp + (size_t)B_ * K_ * H_;

  proj_kernel<<<512, 128, 0, stream>>>(query, key, Wq, Wk, qp, kp);
  score_kernel<<<dim3(K_ / 64, Q_ / 16, B_), 256, 0, stream>>>(qp, kp, Wv, mask, sc);
  softmax_kernel<<<B_ * Q_, 256, 0, stream>>>(sc);
  av_kernel<<<512, 128, 0, stream>>>(sc, value, out);
}